// LSTMLayerBidirectional_46514495816259
// MI455X (gfx1250) — compile-verified
//
#include <hip/hip_runtime.h>

typedef __attribute__((ext_vector_type(2))) float v2f;
typedef __attribute__((ext_vector_type(8))) float v8f;

#define T_LEN 512
#define BATCH 32
#define INP   1024
#define HID   1024
#define G4    4096            // 4*HID
#define MTOT  16384           // T_LEN*BATCH
#define KS    8               // K-split factor in the step kernel
#define KSL   (HID / KS)      // 128

#define WMMA_F32(A, B, C) \
    __builtin_amdgcn_wmma_f32_16x16x4_f32(false, (A), false, (B), (short)0, (C), false, false)

__device__ __forceinline__ float sigmoidf_(float x) {
    return 1.0f / (1.0f + __expf(-x));
}
__device__ __forceinline__ float tanhf_(float x) {
    return 1.0f - 2.0f / (__expf(2.0f * x) + 1.0f);   // saturates to +/-1
}

// ---------------------------------------------------------------------------
// Kernel 1: XG[d][m][g] = sum_k x[m][k] * Wih_d[g][k] + bih_d[g] + bhh_d[g]
// One wave per 32x64 macro-tile (2 M-tiles x 4 N-tiles = 8 accumulators).
// Loads per k-step: 2 A-pairs + 4 B-pairs feed 8 WMMAs (4x A reuse, 2x B).
// waves = 2 dirs * 512 mtg * 64 ntg = 65536 -> 8192 blocks of 8 waves.
// ---------------------------------------------------------------------------
__global__ __launch_bounds__(256) void xg_gemm_kernel(
    const float* __restrict__ x,
    const float* __restrict__ Wih_f, const float* __restrict__ Wih_b,
    const float* __restrict__ bih_f, const float* __restrict__ bhh_f,
    const float* __restrict__ bih_b, const float* __restrict__ bhh_b,
    float* __restrict__ xg)
{
    const int lane = threadIdx.x & 31;
    const int l15  = lane & 15;
    const int lh   = lane >> 4;                    // K-pair half (A and B)
    int w = blockIdx.x * 8 + (threadIdx.x >> 5);   // 0..65535
    const int d   = w >> 15;
    w &= 0x7FFF;
    const int mtg = w >> 6;                        // 0..511 (32 rows each)
    const int ntg = w & 63;                        // 0..63  (64 cols each)

    const float* W  = d ? Wih_b : Wih_f;
    const float* b1 = d ? bih_b : bih_f;
    const float* b2 = d ? bhh_b : bhh_f;

    const int m0 = mtg * 32;
    const int n0 = ntg * 64;

    const float* ap0 = x + (size_t)(m0 + l15)      * INP + 2 * lh;
    const float* ap1 = x + (size_t)(m0 + 16 + l15) * INP + 2 * lh;
    const float* bp0 = W + (size_t)(n0 +  0 + l15) * INP + 2 * lh;
    const float* bp1 = W + (size_t)(n0 + 16 + l15) * INP + 2 * lh;
    const float* bp2 = W + (size_t)(n0 + 32 + l15) * INP + 2 * lh;
    const float* bp3 = W + (size_t)(n0 + 48 + l15) * INP + 2 * lh;

    v8f acc[2][4] = {};
    #pragma unroll 4
    for (int k = 0; k < INP; k += 4) {
        v2f a0 = *(const v2f*)(ap0 + k);
        v2f a1 = *(const v2f*)(ap1 + k);
        v2f b0 = *(const v2f*)(bp0 + k);
        v2f b1v = *(const v2f*)(bp1 + k);
        v2f b2v = *(const v2f*)(bp2 + k);
        v2f b3 = *(const v2f*)(bp3 + k);
        acc[0][0] = WMMA_F32(a0, b0,  acc[0][0]);
        acc[0][1] = WMMA_F32(a0, b1v, acc[0][1]);
        acc[0][2] = WMMA_F32(a0, b2v, acc[0][2]);
        acc[0][3] = WMMA_F32(a0, b3,  acc[0][3]);
        acc[1][0] = WMMA_F32(a1, b0,  acc[1][0]);
        acc[1][1] = WMMA_F32(a1, b1v, acc[1][1]);
        acc[1][2] = WMMA_F32(a1, b2v, acc[1][2]);
        acc[1][3] = WMMA_F32(a1, b3,  acc[1][3]);
    }

    #pragma unroll
    for (int j = 0; j < 4; ++j) {
        const int col = n0 + j * 16 + l15;
        const float bias = b1[col] + b2[col];
        #pragma unroll
        for (int i = 0; i < 2; ++i) {
            const int rbase = m0 + i * 16 + 8 * lh;   // C layout: vgpr r -> M=r(+8)
            float* outp = xg + ((size_t)d * MTOT + rbase) * G4 + col;
            #pragma unroll
            for (int r = 0; r < 8; ++r) {
                __builtin_nontemporal_store(acc[i][j][r] + bias, outp + (size_t)r * G4);
            }
        }
    }
}

// ---------------------------------------------------------------------------
// Kernel 2: one recurrent timestep, both directions.
// Block (8 waves) owns one (dir, 16-col tile) group: all 32 batch rows, all
// 4 gates. Wave ws handles K slice [ws*128, ws*128+128) with 8 accumulators
// (2 batch-tiles x 4 gates); Whh is read exactly once per dir per step.
// Partials reduced via LDS, then the block does the fused c/h update.
// Grid: 2 dirs * 64 col-tiles = 128 blocks.
// ---------------------------------------------------------------------------
__global__ __launch_bounds__(256) void lstm_step_kernel(
    int s,
    const float* __restrict__ xg,
    const float* __restrict__ Whh_f, const float* __restrict__ Whh_b,
    const float* __restrict__ hr_f,  const float* __restrict__ hr_b,
    float* __restrict__ hw_f,        float* __restrict__ hw_b,
    float* __restrict__ c_f,         float* __restrict__ c_b,
    float* __restrict__ out)
{
    __shared__ float red[64 * KS * 32];            // [a(=j*8+r)][ws][lane], 64 KB

    const int tid  = threadIdx.x;
    const int lane = tid & 31;
    const int l15  = lane & 15;
    const int lh   = lane >> 4;
    const int ws   = tid >> 5;                     // K-slice 0..7
    const int d    = blockIdx.x >> 6;
    const int nt   = blockIdx.x & 63;

    const float* W  = d ? Whh_b : Whh_f;
    const float* hr = d ? hr_b  : hr_f;

    const int col = nt * 16 + l15;
    const int k0  = ws * KSL;

    const float* ap0 = hr + (size_t)(l15)      * HID + k0 + 2 * lh;  // batch 0..15
    const float* ap1 = hr + (size_t)(16 + l15) * HID + k0 + 2 * lh;  // batch 16..31
    const float* bp0 = W + (size_t)(0 * HID + col) * HID + k0 + 2 * lh;
    const float* bp1 = W + (size_t)(1 * HID + col) * HID + k0 + 2 * lh;
    const float* bp2 = W + (size_t)(2 * HID + col) * HID + k0 + 2 * lh;
    const float* bp3 = W + (size_t)(3 * HID + col) * HID + k0 + 2 * lh;

    v8f acc[8] = {};                               // j = mt*4 + gate
    #pragma unroll 4
    for (int k = 0; k < KSL; k += 4) {
        v2f a0 = *(const v2f*)(ap0 + k);
        v2f a1 = *(const v2f*)(ap1 + k);
        v2f b0 = *(const v2f*)(bp0 + k);
        v2f b1v = *(const v2f*)(bp1 + k);
        v2f b2v = *(const v2f*)(bp2 + k);
        v2f b3 = *(const v2f*)(bp3 + k);
        acc[0] = WMMA_F32(a0, b0,  acc[0]);
        acc[1] = WMMA_F32(a0, b1v, acc[1]);
        acc[2] = WMMA_F32(a0, b2v, acc[2]);
        acc[3] = WMMA_F32(a0, b3,  acc[3]);
        acc[4] = WMMA_F32(a1, b0,  acc[4]);
        acc[5] = WMMA_F32(a1, b1v, acc[5]);
        acc[6] = WMMA_F32(a1, b2v, acc[6]);
        acc[7] = WMMA_F32(a1, b3,  acc[7]);
    }

    // partials -> LDS (lane-contiguous, conflict-free)
    #pragma unroll
    for (int j = 0; j < 8; ++j) {
        #pragma unroll
        for (int r = 0; r < 8; ++r) {
            red[((j * 8 + r) * KS + ws) * 32 + lane] = acc[j][r];
        }
    }
    __syncthreads();

    // reduce + fused gate math: 32 batch x 16 cols = 512 elements, 2/thread
    float* hw = d ? hw_b : hw_f;
    float* cs = d ? c_b  : c_f;
    const int tm = d ? (T_LEN - 1 - s) : s;
    const size_t xgrow0 = ((size_t)d * MTOT + (size_t)tm * BATCH) * G4;

    #pragma unroll
    for (int e2 = 0; e2 < 2; ++e2) {
        const int e  = tid * 2 + e2;
        const int b  = e >> 4;                      // batch row 0..31
        const int c  = e & 15;                      // col within tile
        const int mt = b >> 4;
        const int m  = b & 15;
        const int lh2 = m >> 3;
        const int r   = m & 7;
        const int ln  = c + 16 * lh2;               // C-layout lane

        float gv[4];
        #pragma unroll
        for (int q = 0; q < 4; ++q) {
            const int a = (mt * 4 + q) * 8 + r;
            float v = 0.0f;
            #pragma unroll
            for (int p = 0; p < KS; ++p) v += red[(a * KS + p) * 32 + ln];
            gv[q] = v;
        }

        const int colq = nt * 16 + c;
        const size_t xb = xgrow0 + (size_t)b * G4 + colq;
        const float gi = sigmoidf_(gv[0] + __builtin_nontemporal_load(xg + xb));
        const float gf = sigmoidf_(gv[1] + __builtin_nontemporal_load(xg + xb + HID));
        const float gg = tanhf_   (gv[2] + __builtin_nontemporal_load(xg + xb + 2 * HID));
        const float go = sigmoidf_(gv[3] + __builtin_nontemporal_load(xg + xb + 3 * HID));

        const size_t ci = (size_t)b * HID + colq;
        const float cnew = gf * cs[ci] + gi * gg;
        cs[ci] = cnew;
        const float hnew = go * tanhf_(cnew);
        hw[ci] = hnew;
        out[((size_t)tm * BATCH + b) * (2 * HID) + (size_t)d * HID + colq] = hnew;
    }
}

// ---------------------------------------------------------------------------
// Host launcher. Workspace: XG (2*MTOT*G4 floats = 536.9 MB) + h ping-pong
// alternates (2*B*H floats). Final states (hf,cf,hb,cb) live in the tail of
// d_out; with T even the last h write lands there.
// ---------------------------------------------------------------------------
extern "C" void kernel_launch(void* const* d_in, const int* in_sizes, int n_in,
                              void* d_out, int out_size, void* d_ws, size_t ws_size,
                              hipStream_t stream) {
    const float* x     = (const float*)d_in[0];
    const float* h0f   = (const float*)d_in[1];
    const float* c0f   = (const float*)d_in[2];
    const float* h0b   = (const float*)d_in[3];
    const float* c0b   = (const float*)d_in[4];
    const float* Wih_f = (const float*)d_in[5];
    const float* Whh_f = (const float*)d_in[6];
    const float* bih_f = (const float*)d_in[7];
    const float* bhh_f = (const float*)d_in[8];
    const float* Wih_b = (const float*)d_in[9];
    const float* Whh_b = (const float*)d_in[10];
    const float* bih_b = (const float*)d_in[11];
    const float* bhh_b = (const float*)d_in[12];

    float* out = (float*)d_out;
    const size_t BH = (size_t)BATCH * HID;                      // 32768
    float* states = out + (size_t)T_LEN * BATCH * 2 * HID;      // hf,cf,hb,cb
    float* hm_f = states;
    float* cm_f = states + BH;
    float* hm_b = states + 2 * BH;
    float* cm_b = states + 3 * BH;

    float* xgbuf = (float*)d_ws;
    float* halt  = (float*)d_ws + (size_t)2 * MTOT * G4;
    float* ha_f  = halt;
    float* ha_b  = halt + BH;

    // initialize recurrent state (deterministic per call)
    hipMemcpyAsync(hm_f, h0f, BH * sizeof(float), hipMemcpyDeviceToDevice, stream);
    hipMemcpyAsync(cm_f, c0f, BH * sizeof(float), hipMemcpyDeviceToDevice, stream);
    hipMemcpyAsync(hm_b, h0b, BH * sizeof(float), hipMemcpyDeviceToDevice, stream);
    hipMemcpyAsync(cm_b, c0b, BH * sizeof(float), hipMemcpyDeviceToDevice, stream);

    // big parallel GEMM: input projections for both directions
    xg_gemm_kernel<<<8192, 256, 0, stream>>>(x, Wih_f, Wih_b,
                                             bih_f, bhh_f, bih_b, bhh_b, xgbuf);

    // sequential recurrence: one fused launch per timestep
    for (int s = 0; s < T_LEN; ++s) {
        const bool even = (s & 1) == 0;
        const float* hrf = even ? hm_f : ha_f;
        const float* hrb = even ? hm_b : ha_b;
        float* hwf = even ? ha_f : hm_f;
        float* hwb = even ? ha_b : hm_b;
        lstm_step_kernel<<<128, 256, 0, stream>>>(s, xgbuf, Whh_f, Whh_b,
                                                  hrf, hrb, hwf, hwb,
                                                  cm_f, cm_b, out);
    }
}